// GCN_91225105367234
// MI455X (gfx1250) — compile-verified
//
#include <hip/hip_runtime.h>

typedef __attribute__((ext_vector_type(2))) float v2f;
typedef __attribute__((ext_vector_type(4))) float v4f;
typedef __attribute__((ext_vector_type(8))) float v8f;

#define B_DIM 8
#define N_DIM 2048
#define F_DIMM 128
#define ROWS (B_DIM * N_DIM)   // 16384
#define SLICES 32              // reduction slices per batch
#define CHUNK (N_DIM / SLICES) // 64 rows per slice

// ---------------------------------------------------------------------------
// Kernel 1: a_i[row] = x[row,:] . adj_w[:128], a_j[row] = x[row,:] . adj_w[128:]
// One wave (32 lanes) per row; float4 per lane; wave32 shuffle reduction.
// ---------------------------------------------------------------------------
__global__ void gcn_row_dots(const float* __restrict__ x,
                             const float* __restrict__ adj_w,
                             float* __restrict__ ai,
                             float* __restrict__ aj) {
    int gid  = blockIdx.x * blockDim.x + threadIdx.x;
    int row  = gid >> 5;
    int lane = gid & 31;
    v4f xv = *(const v4f*)(x + (size_t)row * F_DIMM + lane * 4);
    v4f w1 = *(const v4f*)(adj_w + lane * 4);
    v4f w2 = *(const v4f*)(adj_w + F_DIMM + lane * 4);
    float s1 = xv.x * w1.x + xv.y * w1.y + xv.z * w1.z + xv.w * w1.w;
    float s2 = xv.x * w2.x + xv.y * w2.y + xv.z * w2.z + xv.w * w2.w;
    #pragma unroll
    for (int off = 16; off > 0; off >>= 1) {
        s1 += __shfl_xor(s1, off, 32);
        s2 += __shfl_xor(s2, off, 32);
    }
    if (lane == 0) { ai[row] = s1; aj[row] = s2; }
}

// ---------------------------------------------------------------------------
// Kernel 2: sumAj[b] = sum_n a_j[b,n]   (one block per batch; only 64KB total)
// ---------------------------------------------------------------------------
__global__ void gcn_batch_sum(const float* __restrict__ aj,
                              float* __restrict__ sumAj) {
    __shared__ float sm[256];
    int b = blockIdx.x;
    float s = 0.f;
    for (int n = threadIdx.x; n < N_DIM; n += 256)
        s += aj[(size_t)b * N_DIM + n];
    sm[threadIdx.x] = s;
    __syncthreads();
    #pragma unroll
    for (int off = 128; off > 0; off >>= 1) {
        if (threadIdx.x < (unsigned)off) sm[threadIdx.x] += sm[threadIdx.x + off];
        __syncthreads();
    }
    if (threadIdx.x == 0) sumAj[b] = sm[0];
}

// ---------------------------------------------------------------------------
// Kernel 3: H = x @ weight using fp32 WMMA (V_WMMA_F32_16X16X4_F32).
// Block = 256 threads = 8 waves. blockIdx.x selects the 16-row M tile;
// wave id selects one of 8 16-column N tiles. K=128 -> 32 WMMA steps.
// A 16x4 fp32 layout: lanes 0-15 carry M=lane, (v0,v1)=(K0,K1);
//                     lanes 16-31 carry M=lane-16, (v0,v1)=(K2,K3).
// B 4x16 mirrors it: (v0,v1) = rows (K0+2*khalf, +1) at column lane&15.
// C/D 16x16: VGPR i = row i (lanes 0-15) / row 8+i (lanes 16-31).
// All lanes active, no divergence -> EXEC == all ones as WMMA requires.
// ---------------------------------------------------------------------------
__global__ void gcn_xw_wmma(const float* __restrict__ x,
                            const float* __restrict__ w,
                            float* __restrict__ H) {
    const int lane  = threadIdx.x & 31;
    const int wave  = threadIdx.x >> 5;   // 0..7 -> N tile
    const int khalf = lane >> 4;          // 0 or 1
    const int l15   = lane & 15;
    const int rowA  = blockIdx.x * 16 + l15;
    const int colB  = wave * 16 + l15;

    const float* xrow = x + (size_t)rowA * F_DIMM + 2 * khalf;
    const float* wcol = w + (size_t)(2 * khalf) * F_DIMM + colB;

    v8f c = {0.f, 0.f, 0.f, 0.f, 0.f, 0.f, 0.f, 0.f};
    #pragma unroll
    for (int k0 = 0; k0 < F_DIMM; k0 += 4) {
        v2f a = *(const v2f*)(xrow + k0);             // A[m, k0+2kh], A[m, k0+2kh+1]
        v2f b;
        b.x = wcol[(size_t)k0 * F_DIMM];              // B[k0+2kh,   n]
        b.y = wcol[(size_t)k0 * F_DIMM + F_DIMM];     // B[k0+2kh+1, n]
        c = __builtin_amdgcn_wmma_f32_16x16x4_f32(
                /*neg_a=*/false, a, /*neg_b=*/false, b,
                /*c_mod=*/(short)0, c, /*reuse_a=*/false, /*reuse_b=*/false);
    }

    float* hb = H + (size_t)(blockIdx.x * 16 + 8 * khalf) * F_DIMM + colB;
    #pragma unroll
    for (int i = 0; i < 8; ++i)
        hb[(size_t)i * F_DIMM] = c[i];
}

// ---------------------------------------------------------------------------
// Kernel 4a: sliced per-batch weighted column sums of H (256 blocks -> enough
// concurrency to stream 8MB at HBM rate).
//   P1[b,s,f] = sum_{n in slice s} d[b,n]          * H[b,n,f]
//   P2[b,s,f] = sum_{n in slice s} a_j[b,n]*d[b,n] * H[b,n,f]
// d recomputed on the fly: deg = N*a_i + sumAj + N*b0 + 1; d = max(deg,1)^-1/2.
// Block of 256: tid = f + 128*h, h in {0,1} covers even/odd n within slice.
// ---------------------------------------------------------------------------
__global__ void gcn_s1s2_partial(const float* __restrict__ H,
                                 const float* __restrict__ ai,
                                 const float* __restrict__ aj,
                                 const float* __restrict__ sumAj,
                                 const float* __restrict__ adj_b,
                                 float* __restrict__ P1,
                                 float* __restrict__ P2) {
    __shared__ float sm1[256], sm2[256];
    const int blk = blockIdx.x;
    const int b   = blk / SLICES;
    const int s   = blk % SLICES;
    const int f   = threadIdx.x & 127;
    const int h   = threadIdx.x >> 7;
    const float b0  = adj_b[0];
    const float cst = sumAj[b] + (float)N_DIM * b0 + 1.0f;
    const int n0 = s * CHUNK;

    float s1 = 0.f, s2 = 0.f;
    #pragma unroll 4
    for (int n = n0 + h; n < n0 + CHUNK; n += 2) {
        size_t rn  = (size_t)b * N_DIM + n;
        float aiv  = ai[rn];
        float ajv  = aj[rn];
        float deg  = (float)N_DIM * aiv + cst;
        float d    = 1.0f / sqrtf(fmaxf(deg, 1.0f));
        float hv   = H[rn * F_DIMM + f];
        s1 += d * hv;
        s2 += ajv * d * hv;
    }
    sm1[threadIdx.x] = s1;
    sm2[threadIdx.x] = s2;
    __syncthreads();
    if (h == 0) {
        size_t o = (size_t)blk * F_DIMM + f;
        P1[o] = sm1[f] + sm1[f + 128];
        P2[o] = sm2[f] + sm2[f + 128];
    }
}

// ---------------------------------------------------------------------------
// Kernel 4b: fold the SLICES partials -> S1[b,f], S2[b,f]. 8 blocks x 128.
// ---------------------------------------------------------------------------
__global__ void gcn_s1s2_reduce(const float* __restrict__ P1,
                                const float* __restrict__ P2,
                                float* __restrict__ S1,
                                float* __restrict__ S2) {
    const int b = blockIdx.x;
    const int f = threadIdx.x;
    float s1 = 0.f, s2 = 0.f;
    #pragma unroll
    for (int s = 0; s < SLICES; ++s) {
        size_t o = ((size_t)b * SLICES + s) * F_DIMM + f;
        s1 += P1[o];
        s2 += P2[o];
    }
    S1[(size_t)b * F_DIMM + f] = s1;
    S2[(size_t)b * F_DIMM + f] = s2;
}

// ---------------------------------------------------------------------------
// Kernel 5: in-place finalize on d_out (which currently holds H):
//   out[b,n,f] = relu( d*(a_i+b0)*S1[b,f] + d*S2[b,f] + d*d*H[b,n,f] + bias[f] )
// ---------------------------------------------------------------------------
__global__ void gcn_final(float* __restrict__ out,
                          const float* __restrict__ ai,
                          const float* __restrict__ sumAj,
                          const float* __restrict__ adj_b,
                          const float* __restrict__ S1,
                          const float* __restrict__ S2,
                          const float* __restrict__ bias) {
    size_t idx = (size_t)blockIdx.x * blockDim.x + threadIdx.x;
    int    f   = (int)(idx & 127);
    size_t rn  = idx >> 7;          // = b*N + n
    int    b   = (int)(rn / N_DIM);

    float b0  = adj_b[0];
    float aiv = ai[rn];
    float deg = (float)N_DIM * aiv + sumAj[b] + (float)N_DIM * b0 + 1.0f;
    float d   = 1.0f / sqrtf(fmaxf(deg, 1.0f));

    float hv = out[idx];
    float r  = d * (aiv + b0) * S1[(size_t)b * F_DIMM + f]
             + d * S2[(size_t)b * F_DIMM + f]
             + d * d * hv
             + bias[f];
    out[idx] = fmaxf(r, 0.0f);
}

// ---------------------------------------------------------------------------
extern "C" void kernel_launch(void* const* d_in, const int* in_sizes, int n_in,
                              void* d_out, int out_size, void* d_ws, size_t ws_size,
                              hipStream_t stream) {
    const float* x      = (const float*)d_in[0];   // [8,2048,128]
    const float* adj_w  = (const float*)d_in[1];   // [256]
    const float* adj_b  = (const float*)d_in[2];   // [1]
    const float* weight = (const float*)d_in[3];   // [128,128]
    const float* bias   = (const float*)d_in[4];   // [128]
    float* out = (float*)d_out;                    // [8,2048,128]

    // Workspace layout (floats):
    //   ai[16384] aj[16384] sumAj[16] S1[1024] S2[1024]
    //   P1[8*32*128=32768] P2[32768]           total ~400KB
    float* ws    = (float*)d_ws;
    float* ai    = ws;
    float* aj    = ws + ROWS;
    float* sumAj = ws + 2 * ROWS;
    float* S1    = ws + 2 * ROWS + 16;
    float* S2    = S1 + B_DIM * F_DIMM;
    float* P1    = S2 + B_DIM * F_DIMM;
    float* P2    = P1 + B_DIM * SLICES * F_DIMM;

    // 1) row dots: 16384 rows, 1 wave each
    gcn_row_dots<<<(ROWS * 32) / 256, 256, 0, stream>>>(x, adj_w, ai, aj);
    // 2) per-batch sum of a_j
    gcn_batch_sum<<<B_DIM, 256, 0, stream>>>(aj, sumAj);
    // 3) H = x @ W  (fp32 WMMA), H staged in d_out
    gcn_xw_wmma<<<ROWS / 16, 256, 0, stream>>>(x, weight, out);
    // 4a) sliced S1/S2 partials (256 blocks -> HBM-rate streaming of H)
    gcn_s1s2_partial<<<B_DIM * SLICES, 256, 0, stream>>>(out, ai, aj, sumAj,
                                                         adj_b, P1, P2);
    // 4b) fold partials
    gcn_s1s2_reduce<<<B_DIM, F_DIMM, 0, stream>>>(P1, P2, S1, S2);
    // 5) finalize in place
    gcn_final<<<(ROWS * F_DIMM) / 256, 256, 0, stream>>>(out, ai, sumAj, adj_b,
                                                         S1, S2, bias);
}